// BlockWiseEmbedding_83708912599528
// MI455X (gfx1250) — compile-verified
//
#include <hip/hip_runtime.h>

#define TOKENS   16384   // B*L = 8*2048
#define ODIM     512
#define NBLOCKS  4
#define TBL_ROWS 25000
#define TILE_M   16
#define OSTRIDE  (ODIM + 4)   // LDS out-tile row stride (floats), kills bank conflicts

typedef __attribute__((ext_vector_type(2))) float v2f;
typedef __attribute__((ext_vector_type(8))) float v8f;

// packed-trans float offsets per block (cumulative S_b * 512)
__constant__ int kPackOffHost[NBLOCKS] = {0, 32768, 98304, 229376};
#define PACK_TOTAL_FLOATS 491520   // (64+128+256+512)*512

// ---------------- kernel 0: zero per-block counters ----------------
__global__ void zero_counts_kernel(int* __restrict__ counts) {
    if (threadIdx.x < NBLOCKS) counts[threadIdx.x] = 0;
}

// ---------------- kernel 1: wave-aggregated token binning ----------------
__global__ void bin_tokens_kernel(const int* __restrict__ src,
                                  const int* __restrict__ block_assignment,
                                  const int* __restrict__ local_assignment,
                                  int* __restrict__ counts,
                                  int* __restrict__ tok_list,
                                  int* __restrict__ la_list) {
    const int t = blockIdx.x * blockDim.x + threadIdx.x;
    if (t >= TOKENS) return;                    // TOKENS % 256 == 0: waves stay full
    const int lane = threadIdx.x & 31;
    const int s  = src[t];
    const int ba = block_assignment[s];
    int la = local_assignment[s];
    la = la < 0 ? 0 : (la > (TBL_ROWS - 1) ? (TBL_ROWS - 1) : la);

    #pragma unroll
    for (int b = 0; b < NBLOCKS; ++b) {
        unsigned long long m = __ballot(ba == b);   // wave32: low 32 bits
        if (ba == b) {
            const int leader = __ffsll((long long)m) - 1;
            const int cnt    = __popcll(m);
            const int myoff  = __popcll(m & ((1ull << lane) - 1ull));
            int base = 0;
            if (lane == leader) base = atomicAdd(&counts[b], cnt);
            base = __shfl(base, leader);
            tok_list[b * TOKENS + base + myoff] = t;
            la_list[b * TOKENS + base + myoff]  = la;
        }
    }
}

// ---------------- kernel 2: pack trans into (K/2, N, 2) pair layout ----------------
// packed[p*1024 + 2*n + {0,1}] = trans[(2p)*512 + n], trans[(2p+1)*512 + n]
__global__ void pack_trans_kernel(const float* __restrict__ trans,
                                  float* __restrict__ packed, int S) {
    const int idx = blockIdx.x * blockDim.x + threadIdx.x;
    const int total = S * ODIM / 2;
    if (idx >= total) return;
    const int p = idx >> 9;          // pair index
    const int n = idx & (ODIM - 1);
    float2 v;
    v.x = trans[(size_t)(2 * p)     * ODIM + n];
    v.y = trans[(size_t)(2 * p + 1) * ODIM + n];
    *(float2*)(packed + (size_t)p * (2 * ODIM) + 2 * n) = v;
}

// ---------------- kernel 3: per-block tile GEMM via FP32 WMMA ----------------
// One workgroup (256 threads = 8 waves, wave32) handles 16 tokens x 512 outputs.
// Each wave owns 4 N-tiles (ntile = wave + 8*j); K advances by 4 per WMMA.
template<int S, bool PACKED>
__global__ __launch_bounds__(256)
void gemm_block_kernel(const float* __restrict__ table,
                       const float* __restrict__ trans,
                       const float* __restrict__ packedB,
                       const int*  __restrict__ count_p,
                       const int*  __restrict__ tok_list,
                       const int*  __restrict__ la_list,
                       float*      __restrict__ out) {
    constexpr int LSTRIDE = S + 2;                 // LDS A row pad
    __shared__ float aLds[TILE_M * LSTRIDE];
    __shared__ float oLds[TILE_M * OSTRIDE];
    __shared__ int   toks[TILE_M];

    const int count = *count_p;
    const int tile0 = blockIdx.x * TILE_M;
    if (tile0 >= count) return;                    // uniform across workgroup
    const int nTok = min(TILE_M, count - tile0);

    const int tid = threadIdx.x;
    if (tid < TILE_M)
        toks[tid] = (tid < nTok) ? tok_list[tile0 + tid] : 0;

    // Gather 16 embedding rows (S floats each) into LDS, float4 granularity.
    constexpr int V4_PER_ROW = S / 4;
    constexpr int TOTAL_V4   = TILE_M * V4_PER_ROW;
    for (int i = tid; i < TOTAL_V4; i += 256) {
        const int row = i / V4_PER_ROW;
        const int c4  = i % V4_PER_ROW;
        float4 v = make_float4(0.f, 0.f, 0.f, 0.f);
        if (row < nTok) {
            const int la = la_list[tile0 + row];
            v = *(const float4*)(table + (size_t)la * S + (size_t)c4 * 4);
        }
        float* dst = aLds + row * LSTRIDE + c4 * 4;
        dst[0] = v.x; dst[1] = v.y; dst[2] = v.z; dst[3] = v.w;
    }
    __syncthreads();

    const int lane = tid & 31;
    const int wave = tid >> 5;
    const int m16  = lane & 15;                    // M (for A) / N (for B,D) sub-index
    const int kh   = (lane >> 4) << 1;             // lanes 16-31 carry K+2/K+3

    v8f acc[4];
    #pragma unroll
    for (int j = 0; j < 4; ++j) acc[j] = (v8f){0.f,0.f,0.f,0.f,0.f,0.f,0.f,0.f};

    for (int k = 0; k < S; k += 4) {
        const int kk = k + kh;                     // even
        // A 16x4 tile: VGPR0 = A[m][kk], VGPR1 = A[m][kk+1]  (ISA layout)
        const v2f a = *(const v2f*)(aLds + m16 * LSTRIDE + kk);
        #pragma unroll
        for (int j = 0; j < 4; ++j) {
            const int n = ((wave + 8 * j) << 4) + m16;
            v2f bm;
            if constexpr (PACKED) {
                // single coalesced b64: pair (trans[kk][n], trans[kk+1][n])
                bm = *(const v2f*)(packedB + (size_t)(kk >> 1) * (2 * ODIM) + 2 * n);
            } else {
                bm.x = trans[(size_t)kk * ODIM + n];
                bm.y = trans[(size_t)(kk + 1) * ODIM + n];
            }
            acc[j] = __builtin_amdgcn_wmma_f32_16x16x4_f32(
                false, a, false, bm, (short)0, acc[j], false, false);
        }
    }

    // Stage D tile in LDS, then coalesced b128 row stores.
    // D layout: VGPR r -> M = r + 8*(lane>=16), N = lane&15
    const int mBase = (lane >> 4) << 3;
    #pragma unroll
    for (int j = 0; j < 4; ++j) {
        const int n = ((wave + 8 * j) << 4) + m16;
        #pragma unroll
        for (int r = 0; r < 8; ++r)
            oLds[(mBase + r) * OSTRIDE + n] = acc[j][r];
    }
    __syncthreads();

    constexpr int OV4 = ODIM / 4;                  // 128 float4 per row
    for (int i = tid; i < TILE_M * OV4; i += 256) {
        const int row = i >> 7;
        const int c4  = i & (OV4 - 1);
        if (row < nTok) {
            const float* sp = oLds + row * OSTRIDE + c4 * 4;
            float4 v = make_float4(sp[0], sp[1], sp[2], sp[3]);
            *(float4*)(out + (size_t)toks[row] * ODIM + (size_t)c4 * 4) = v;
        }
    }
}

template<int S>
static void launch_gemm(bool packed, int maxTiles, hipStream_t stream,
                        const float* tbl, const float* trs, const float* pk,
                        const int* cnt, const int* tl, const int* ll, float* out) {
    if (packed)
        gemm_block_kernel<S, true ><<<maxTiles, 256, 0, stream>>>(tbl, trs, pk, cnt, tl, ll, out);
    else
        gemm_block_kernel<S, false><<<maxTiles, 256, 0, stream>>>(tbl, trs, pk, cnt, tl, ll, out);
}

extern "C" void kernel_launch(void* const* d_in, const int* in_sizes, int n_in,
                              void* d_out, int out_size, void* d_ws, size_t ws_size,
                              hipStream_t stream) {
    const int*   src = (const int*)d_in[0];
    const int*   ba  = (const int*)d_in[1];
    const int*   la  = (const int*)d_in[2];
    const float* blk[NBLOCKS] = {(const float*)d_in[3], (const float*)d_in[4],
                                 (const float*)d_in[5], (const float*)d_in[6]};
    const float* trs[NBLOCKS] = {(const float*)d_in[7], (const float*)d_in[8],
                                 (const float*)d_in[9], (const float*)d_in[10]};
    float* out = (float*)d_out;

    // workspace: [counts(4) | pad to 64B | tok_list[4][TOKENS] | la_list[4][TOKENS] | packedB]
    int* counts   = (int*)d_ws;
    int* tok_list = (int*)((char*)d_ws + 64);
    int* la_list  = tok_list + NBLOCKS * TOKENS;
    float* packedB = (float*)(la_list + NBLOCKS * TOKENS);
    const size_t needPacked = 64 + (size_t)2 * NBLOCKS * TOKENS * 4 + (size_t)PACK_TOTAL_FLOATS * 4;
    const bool usePacked = ws_size >= needPacked;

    const int packOff[NBLOCKS] = {0, 32768, 98304, 229376};
    const int Ssize[NBLOCKS]   = {64, 128, 256, 512};

    zero_counts_kernel<<<1, 32, 0, stream>>>(counts);
    bin_tokens_kernel<<<TOKENS / 256, 256, 0, stream>>>(src, ba, la, counts, tok_list, la_list);

    if (usePacked) {
        for (int b = 0; b < NBLOCKS; ++b) {
            const int total = Ssize[b] * ODIM / 2;
            pack_trans_kernel<<<(total + 255) / 256, 256, 0, stream>>>(
                trs[b], packedB + packOff[b], Ssize[b]);
        }
    }

    const int maxTiles = TOKENS / TILE_M;   // over-provisioned; uniform early-exit per tile
    launch_gemm< 64>(usePacked, maxTiles, stream, blk[0], trs[0], packedB + packOff[0],
                     counts + 0, tok_list + 0 * TOKENS, la_list + 0 * TOKENS, out);
    launch_gemm<128>(usePacked, maxTiles, stream, blk[1], trs[1], packedB + packOff[1],
                     counts + 1, tok_list + 1 * TOKENS, la_list + 1 * TOKENS, out);
    launch_gemm<256>(usePacked, maxTiles, stream, blk[2], trs[2], packedB + packOff[2],
                     counts + 2, tok_list + 2 * TOKENS, la_list + 2 * TOKENS, out);
    launch_gemm<512>(usePacked, maxTiles, stream, blk[3], trs[3], packedB + packOff[3],
                     counts + 3, tok_list + 3 * TOKENS, la_list + 3 * TOKENS, out);
}